// MLPDecoder_30313879175767
// MI455X (gfx1250) — compile-verified
//
#include <hip/hip_runtime.h>

// CDNA5 / gfx1250: wave32, WMMA (V_WMMA_F32_16X16X4_F32 for full-f32 matrix math).

typedef __attribute__((ext_vector_type(2))) float v2f;
typedef __attribute__((ext_vector_type(4))) float v4f;
typedef __attribute__((ext_vector_type(8))) float v8f;

#define D_FEAT 128

// One wave32 processes 16 edges. A-fragment (16x4 f32, documented layout):
//   lane l: M = l & 15, K = {0,1} for lanes 0-15, K = {2,3} for lanes 16-31.
// We fold w into A (a = |r-c| * w) and use B = ones so every column of the
// 16x16 f32 accumulator equals the running dot product -> correctness does
// not depend on the f32 B-fragment K mapping.
__global__ __launch_bounds__(256) void edge_absdiff_dot_wmma(
    const float* __restrict__ node,
    const int*   __restrict__ ridx,
    const int*   __restrict__ cidx,
    const float* __restrict__ w,
    const float* __restrict__ bias,
    float*       __restrict__ out,
    int nEdges)
{
    const int lane  = threadIdx.x & 31;
    const int wave  = blockIdx.x * (blockDim.x >> 5) + (threadIdx.x >> 5);
    const int nWave = gridDim.x * (blockDim.x >> 5);
    const int half  = lane >> 4;    // 0: lanes 0-15 (K=0,1), 1: lanes 16-31 (K=2,3)
    const int slot  = lane & 15;    // edge slot within the 16-edge tile

    const float b0  = bias[0];
    const v2f ones  = {1.0f, 1.0f};
    const int nTiles = nEdges >> 4; // 16 edges per tile

    for (int tile = wave; tile < nTiles; tile += nWave) {
        const int ebase = tile << 4;

        // Prefetch next tile's indices (emits global_prefetch on gfx1250).
        const int nb = (tile + nWave) << 4;
        if (nb < nEdges) {
            __builtin_prefetch(ridx + nb + slot, 0, 3);
            __builtin_prefetch(cidx + nb + slot, 0, 3);
        }

        const int ri = ridx[ebase + slot];
        const int ci = cidx[ebase + slot];
        const float* rp = node + (size_t)ri * D_FEAT;
        const float* cp = node + (size_t)ci * D_FEAT;

        v8f acc = {0.f, 0.f, 0.f, 0.f, 0.f, 0.f, 0.f, 0.f};

        #pragma unroll 4
        for (int k = 0; k < D_FEAT; k += 4) {
            const int f = k + (half << 1);      // this lane's two features
            v2f r2 = *(const v2f*)(rp + f);     // global_load_b64, 8B aligned
            v2f c2 = *(const v2f*)(cp + f);
            v2f w2 = *(const v2f*)(w + f);      // tiny, L0-resident
            v2f a;
            a.x = __builtin_fabsf(r2.x - c2.x) * w2.x;   // |d| folds into VOP3 abs
            a.y = __builtin_fabsf(r2.y - c2.y) * w2.y;
            // D = A(16x4) * B(4x16, ones) + C : accumulates dot in all 16 cols
            acc = __builtin_amdgcn_wmma_f32_16x16x4_f32(
                false, a, false, ones, (short)0, acc, false, false);
        }

        // C/D f32 16x16 layout: lane l holds column N=l&15; VGPR v holds row
        // M = v + 8*(l>>4). All columns identical -> lanes 0 and 16 each store
        // 8 contiguous results (edges M=0..7 and M=8..15).
        if (slot == 0) {
            float* o = out + ebase + (half << 3);
            v4f lo = { acc[0] + b0, acc[1] + b0, acc[2] + b0, acc[3] + b0 };
            v4f hi = { acc[4] + b0, acc[5] + b0, acc[6] + b0, acc[7] + b0 };
            *(v4f*)(o)     = lo;
            *(v4f*)(o + 4) = hi;
        }
    }
}

// Scalar fallback for a non-multiple-of-16 edge tail (not hit for 600000,
// but kept for generality; launched only when needed — deterministic).
__global__ void edge_absdiff_dot_tail(
    const float* __restrict__ node,
    const int*   __restrict__ ridx,
    const int*   __restrict__ cidx,
    const float* __restrict__ w,
    const float* __restrict__ bias,
    float*       __restrict__ out,
    int start, int nEdges)
{
    int e = start + blockIdx.x * blockDim.x + threadIdx.x;
    if (e >= nEdges) return;
    const float* rp = node + (size_t)ridx[e] * D_FEAT;
    const float* cp = node + (size_t)cidx[e] * D_FEAT;
    float acc = 0.f;
    #pragma unroll 8
    for (int k = 0; k < D_FEAT; ++k)
        acc += __builtin_fabsf(rp[k] - cp[k]) * w[k];
    out[e] = acc + bias[0];
}

extern "C" void kernel_launch(void* const* d_in, const int* in_sizes, int n_in,
                              void* d_out, int out_size, void* d_ws, size_t ws_size,
                              hipStream_t stream)
{
    (void)n_in; (void)d_ws; (void)ws_size;
    const float* node = (const float*)d_in[0];
    const int*   ridx = (const int*)d_in[1];
    const int*   cidx = (const int*)d_in[2];
    const float* w    = (const float*)d_in[3];
    const float* b    = (const float*)d_in[4];
    float*       out  = (float*)d_out;

    const int nEdges = in_sizes[1];        // 600000
    const int nTiles = nEdges >> 4;        // 16 edges per wave-tile
    const int wavesPerBlock = 256 / 32;    // 8 waves per block (wave32)
    int blocks = (nTiles + wavesPerBlock - 1) / wavesPerBlock;
    if (blocks < 1) blocks = 1;

    edge_absdiff_dot_wmma<<<blocks, 256, 0, stream>>>(
        node, ridx, cidx, w, b, out, nEdges);

    const int tail = nEdges & 15;
    if (tail) {
        edge_absdiff_dot_tail<<<1, 32, 0, stream>>>(
            node, ridx, cidx, w, b, out, nEdges - tail, nEdges);
    }
    (void)out_size;
}